// DownSample_46136538694254
// MI455X (gfx1250) — compile-verified
//
#include <hip/hip_runtime.h>

typedef __attribute__((ext_vector_type(2))) float v2f;
typedef __attribute__((ext_vector_type(8))) float v8f;

#define KNN    16
#define DPITCH 36   // dtile row pitch (floats): rows 16B-aligned; store phase conflict-free
#define MPITCH 33   // merge array pitch: conflict-free per-row scans

// Sorted (ascending) top-16 in VGPRs. Insert = 16 v_cmp masks + cndmask shift
// network; no SALU chains, no rescan (new worst is bd[15]).
__device__ __forceinline__ void topk_insert(float (&bd)[KNN], int (&bi)[KNN],
                                            float &worst, float v, int idx)
{
    if (v < worst) {                        // strict: ties keep earlier index
        bool gt[KNN];
#pragma unroll
        for (int i = 0; i < KNN; ++i) gt[i] = bd[i] > v;
#pragma unroll
        for (int i = KNN - 1; i > 0; --i) { // top-down: in-place safe
            bd[i] = gt[i - 1] ? bd[i - 1] : (gt[i] ? v   : bd[i]);
            bi[i] = gt[i - 1] ? bi[i - 1] : (gt[i] ? idx : bi[i]);
        }
        bd[0] = gt[0] ? v   : bd[0];
        bi[0] = gt[0] ? idx : bi[0];
        worst = bd[KNN - 1];
    }
}

// Guarded 4-candidate insert: one divergent branch per quad (quad-min test).
__device__ __forceinline__ void topk_insert4(float (&bd)[KNN], int (&bi)[KNN],
                                             float &worst, const float4 v, int idx0)
{
    const float mq = fminf(fminf(v.x, v.y), fminf(v.z, v.w));
    if (mq < worst) {
        topk_insert(bd, bi, worst, v.x, idx0 + 0);
        topk_insert(bd, bi, worst, v.y, idx0 + 1);
        topk_insert(bd, bi, worst, v.z, idx0 + 2);
        topk_insert(bd, bi, worst, v.w, idx0 + 3);
    }
}

// One wave per 16-query tile. Distance ranking via V_WMMA_F32_16X16X4_F32:
//   A[m][:] = (-2qx, -2qy, -2qz, 1), B[:][n] = (px, py, pz, ||p||^2)
//   D[m][n] = d(m,n) - ||q_m||^2  (row-constant shift: top-k invariant)
__global__ __launch_bounds__(32)
void knn_downsample_kernel(const float* __restrict__ p1,   // (B,N,3)
                           const float* __restrict__ x1,   // (B,C,N)
                           const float* __restrict__ p2,   // (B,M,3)
                           float* __restrict__ out,        // (B,C,M)
                           int B, int N, int M, int C)
{
    __shared__ float dtile[16][DPITCH];   // 32 distance columns = 2 WMMA tiles
    __shared__ float mergd[16][MPITCH];
    __shared__ int   mergi[16][MPITCH];
    __shared__ int   topis[16][16];

    const int lane = threadIdx.x;
    const int row  = lane & 15;
    const int half = lane >> 4;           // 0 or 1
    const int tilesPerBatch = M / 16;
    const int b  = blockIdx.x / tilesPerBatch;
    const int m0 = (blockIdx.x % tilesPerBatch) * 16;

    // ---- A operand (16x4 f32): lanes 0-15 hold K0,K1; lanes 16-31 hold K2,K3 ----
    const float* q = p2 + ((size_t)b * M + (size_t)(m0 + row)) * 3;
    const float qx = q[0], qy = q[1], qz = q[2];
    v2f amat;
    amat.x = half ? (-2.0f * qz) : (-2.0f * qx);
    amat.y = half ? 1.0f         : (-2.0f * qy);

    // ---- per-lane sorted top-16 over (row = lane&15, this lane's column half) ----
    float bd[KNN]; int bi[KNN];
#pragma unroll
    for (int i = 0; i < KNN; ++i) { bd[i] = __builtin_inff(); bi[i] = 0; }
    float worst = __builtin_inff();

    const float* pbase = p1 + (size_t)b * N * 3;
    const int jbase = half * 8;
    const int rbase = half * 8;

    for (int n0 = 0; n0 < N; n0 += 32) {
        // ---- B operands for two 16-point tiles (4x16 f32 layout, mirror of A) ----
        const float* pt0 = pbase + (size_t)(n0 + row) * 3;
        const float* pt1 = pbase + (size_t)(n0 + 16 + row) * 3;
        const float p0x = pt0[0], p0y = pt0[1], p0z = pt0[2];
        const float p1x = pt1[0], p1y = pt1[1], p1z = pt1[2];
        const float nr0 = p0x * p0x + p0y * p0y + p0z * p0z;
        const float nr1 = p1x * p1x + p1y * p1y + p1z * p1z;
        v2f bm0, bm1;
        bm0.x = half ? p0z : p0x;  bm0.y = half ? nr0 : p0y;
        bm1.x = half ? p1z : p1x;  bm1.y = half ? nr1 : p1y;

        v8f cz = {0.f, 0.f, 0.f, 0.f, 0.f, 0.f, 0.f, 0.f};
        v8f d0 = __builtin_amdgcn_wmma_f32_16x16x4_f32(false, amat, false, bm0,
                                                       (short)0, cz, false, false);
        v8f d1 = __builtin_amdgcn_wmma_f32_16x16x4_f32(false, amat, false, bm1,
                                                       (short)0, cz, false, false);

        __syncthreads();
        // C/D layout: VGPR r -> row r + 8*half, col = lane&15
#pragma unroll
        for (int r = 0; r < 8; ++r) {
            dtile[rbase + r][row]      = d0[r];
            dtile[rbase + r][16 + row] = d1[r];
        }
        __syncthreads();

        // ---- selection: this lane scans 8 cols of each tile for its row ----
        const float4* t0 = reinterpret_cast<const float4*>(&dtile[row][jbase]);
        const float4* t1 = reinterpret_cast<const float4*>(&dtile[row][16 + jbase]);
        const float4 a0 = t0[0], a1 = t0[1];
        const float4 c0 = t1[0], c1 = t1[1];

        topk_insert4(bd, bi, worst, a0, n0 + jbase);
        topk_insert4(bd, bi, worst, a1, n0 + jbase + 4);
        topk_insert4(bd, bi, worst, c0, n0 + 16 + jbase);
        topk_insert4(bd, bi, worst, c1, n0 + 16 + jbase + 4);
    }

    // ---- merge the two per-row half-lists -> final top-16 indices ----
#pragma unroll
    for (int k = 0; k < KNN; ++k) {
        mergd[row][half * KNN + k] = bd[k];
        mergi[row][half * KNN + k] = bi[k];
    }
    __syncthreads();

    if (lane < 16) {
        float fd[KNN]; int fi[KNN];
#pragma unroll
        for (int i = 0; i < KNN; ++i) { fd[i] = __builtin_inff(); fi[i] = 0; }
        float w2 = __builtin_inff();
        for (int k = 0; k < 2 * KNN; ++k)
            topk_insert(fd, fi, w2, mergd[lane][k], mergi[lane][k]);
#pragma unroll
        for (int k = 0; k < KNN; ++k) topis[lane][k] = fi[k];
    }
    __syncthreads();

    // ---- gather + mean: out[b][c][m0+m] = (1/16) sum_k x1[b][c][idx] ----
    const float* feat = x1  + (size_t)b * C * N;
    float*       outb = out + (size_t)b * C * M;
    const float inv = 1.0f / (float)KNN;

    for (int m = 0; m < 16; ++m) {
        for (int c = lane; c < C; c += 32) {
            const float* fc = feat + (size_t)c * N;
            float acc = 0.f;
#pragma unroll
            for (int k = 0; k < KNN; ++k)
                acc += fc[topis[m][k]];
            outb[(size_t)c * M + (size_t)(m0 + m)] = acc * inv;
        }
    }
}

extern "C" void kernel_launch(void* const* d_in, const int* in_sizes, int n_in,
                              void* d_out, int out_size, void* d_ws, size_t ws_size,
                              hipStream_t stream)
{
    const float* p1 = (const float*)d_in[0];   // (B,N,3)
    const float* x1 = (const float*)d_in[1];   // (B,C,N)
    const float* p2 = (const float*)d_in[2];   // (B,M,3)
    float*       out = (float*)d_out;          // (B,C,M)

    const int B = 2;                            // fixed by reference
    const int N = in_sizes[0] / (B * 3);        // 16384
    const int M = in_sizes[2] / (B * 3);        // 4096
    const int C = in_sizes[1] / (B * N);        // 64

    const int blocks = B * (M / 16);            // 512 one-wave workgroups
    knn_downsample_kernel<<<blocks, 32, 0, stream>>>(p1, x1, p2, out, B, N, M, C);
}